// SimpleGIN_20538533610166
// MI455X (gfx1250) — compile-verified
//
#include <hip/hip_runtime.h>
#include <hip/hip_bf16.h>

// ---------------------------------------------------------------------------
// GIN forward on gfx1250 (MI455X), fp32 end-to-end.
//   agg1 = x + scatter_add(x[src] -> dst)            [N,64]
//   h    = relu( relu(agg1@W1+b1) @ W2 + b2 )        [N,32]
//   agg2 = h + scatter_add(h[src] -> dst)            [N,32]
//   out  = relu( agg2 @ W3 + b3 )                    [N,64]
// GEMMs use V_WMMA_F32_16X16X4_F32 (one 16-node tile per wave32).
// ---------------------------------------------------------------------------

#define GN_NODES   100000
#define GN_EDGES   1600000
#define GIN_DIM    64
#define GHID       32
#define GOUT_DIM   64
#define GN_TILES   (GN_NODES / 16)   // 6250, exact

typedef __attribute__((ext_vector_type(2))) float v2f;
typedef __attribute__((ext_vector_type(8))) float v8f;

__device__ __forceinline__ void atomic_add_f32(float* p, float v) {
    (void)__hip_atomic_fetch_add(p, v, __ATOMIC_RELAXED, __HIP_MEMORY_SCOPE_AGENT);
}

// ---------------------------------------------------------------------------
// Kernel 1: agg1 = x   (float4 copy, N*64 floats)
// ---------------------------------------------------------------------------
__global__ __launch_bounds__(256) void gin_copy64(const float* __restrict__ x,
                                                  float* __restrict__ agg1) {
    int i = blockIdx.x * 256 + threadIdx.x;           // 1.6M float4 groups
    if (i >= GN_NODES * GIN_DIM / 4) return;
    const float4* src = (const float4*)x;
    float4*       dst = (float4*)agg1;
    dst[i] = src[i];
}

// ---------------------------------------------------------------------------
// Kernel 2: scatter-add x[src] into agg1[dst], dim 64. 16 threads per edge,
// 4 floats each. Gathers/atomics are L2-resident (25.6 MB << 192 MB L2).
// ---------------------------------------------------------------------------
__global__ __launch_bounds__(256) void gin_scatter64(const float* __restrict__ x,
                                                     const int* __restrict__ src,
                                                     const int* __restrict__ dst,
                                                     float* __restrict__ agg1) {
    int t = blockIdx.x * 256 + threadIdx.x;           // E*16 = 25.6M threads
    int e = t >> 4;
    int g = t & 15;
    if (e >= GN_EDGES) return;
    int s = src[e], d = dst[e];
    float4 v = *(const float4*)(x + (size_t)s * GIN_DIM + g * 4);
    float* p = agg1 + (size_t)d * GIN_DIM + g * 4;
    atomic_add_f32(p + 0, v.x);
    atomic_add_f32(p + 1, v.y);
    atomic_add_f32(p + 2, v.z);
    atomic_add_f32(p + 3, v.w);
}

// ---------------------------------------------------------------------------
// Kernel 3: h = relu( relu(agg1@W1+b1) @ W2 + b2 ); agg2 = h.
// One wave32 per 16-node tile; 8 waves per block.
// WMMA f32 16x16x4 fragment mapping (wave32):
//   A : lane l, vgpr v -> M = l%16, K = v + 2*(l/16)   (v2f = contiguous K pair)
//   B : lane l, vgpr v -> N = l%16, K = v + 2*(l/16)
//   C/D: lane l, vgpr r -> M = r + 8*(l/16), N = l%16
// Intermediate ReLU tile staged via per-wave LDS to re-layout D -> A.
// ---------------------------------------------------------------------------
__global__ __launch_bounds__(256) void gin_mlp1(const float* __restrict__ agg1,
                                                const float* __restrict__ W1,
                                                const float* __restrict__ b1,
                                                const float* __restrict__ W2,
                                                const float* __restrict__ b2,
                                                float* __restrict__ h,
                                                float* __restrict__ agg2) {
    __shared__ float lds[8][16][GHID + 1];            // per-wave 16x32 tile (+pad)
    const int wave = threadIdx.x >> 5;
    const int lane = threadIdx.x & 31;
    const int tile = blockIdx.x * 8 + wave;
    if (tile >= GN_TILES) return;                     // wave-uniform: EXEC stays full
    const int row0 = tile * 16;
    const int m    = lane & 15;                       // A row / C col index base
    const int half = lane >> 4;                       // 0 or 1
    const int n    = lane & 15;

    // ---- GEMM1: [16x64] @ [64x32] + b1, ReLU -> LDS ----
    const float* arow = agg1 + (size_t)(row0 + m) * GIN_DIM;
    for (int nt = 0; nt < 2; ++nt) {
        const int col = nt * 16 + n;
        float bias = b1[col];
        v8f c;
        for (int r = 0; r < 8; ++r) c[r] = bias;
        for (int k = 0; k < GIN_DIM; k += 4) {
            const int ka = k + 2 * half;
            v2f a, b;
            a[0] = arow[ka];
            a[1] = arow[ka + 1];
            b[0] = W1[(size_t)ka * GHID + col];
            b[1] = W1[(size_t)(ka + 1) * GHID + col];
            c = __builtin_amdgcn_wmma_f32_16x16x4_f32(false, a, false, b,
                                                      (short)0, c, false, false);
        }
        for (int r = 0; r < 8; ++r)
            lds[wave][r + 8 * half][col] = fmaxf(c[r], 0.0f);
    }

    // ---- GEMM2: [16x32] @ [32x32] + b2, ReLU -> h, agg2 ----
    for (int nt = 0; nt < 2; ++nt) {
        const int col = nt * 16 + n;
        float bias = b2[col];
        v8f c;
        for (int r = 0; r < 8; ++r) c[r] = bias;
        for (int k = 0; k < GHID; k += 4) {
            const int ka = k + 2 * half;
            v2f a, b;
            a[0] = lds[wave][m][ka];
            a[1] = lds[wave][m][ka + 1];
            b[0] = W2[(size_t)ka * GHID + col];
            b[1] = W2[(size_t)(ka + 1) * GHID + col];
            c = __builtin_amdgcn_wmma_f32_16x16x4_f32(false, a, false, b,
                                                      (short)0, c, false, false);
        }
        for (int r = 0; r < 8; ++r) {
            float v = fmaxf(c[r], 0.0f);              // inter-layer ReLU
            size_t idx = (size_t)(row0 + r + 8 * half) * GHID + col;
            h[idx]    = v;
            agg2[idx] = v;                            // seed agg2 = h
        }
    }
}

// ---------------------------------------------------------------------------
// Kernel 4: scatter-add h[src] into agg2[dst], dim 32. 8 threads per edge.
// ---------------------------------------------------------------------------
__global__ __launch_bounds__(256) void gin_scatter32(const float* __restrict__ h,
                                                     const int* __restrict__ src,
                                                     const int* __restrict__ dst,
                                                     float* __restrict__ agg2) {
    int t = blockIdx.x * 256 + threadIdx.x;           // E*8 = 12.8M threads
    int e = t >> 3;
    int g = t & 7;
    if (e >= GN_EDGES) return;
    int s = src[e], d = dst[e];
    float4 v = *(const float4*)(h + (size_t)s * GHID + g * 4);
    float* p = agg2 + (size_t)d * GHID + g * 4;
    atomic_add_f32(p + 0, v.x);
    atomic_add_f32(p + 1, v.y);
    atomic_add_f32(p + 2, v.z);
    atomic_add_f32(p + 3, v.w);
}

// ---------------------------------------------------------------------------
// Kernel 5: out = relu( agg2 @ W3 + b3 )   [16x32]@[32x64] per wave tile.
// ---------------------------------------------------------------------------
__global__ __launch_bounds__(256) void gin_mlp2(const float* __restrict__ agg2,
                                                const float* __restrict__ W3,
                                                const float* __restrict__ b3,
                                                float* __restrict__ out) {
    const int wave = threadIdx.x >> 5;
    const int lane = threadIdx.x & 31;
    const int tile = blockIdx.x * 8 + wave;
    if (tile >= GN_TILES) return;
    const int row0 = tile * 16;
    const int m    = lane & 15;
    const int half = lane >> 4;
    const int n    = lane & 15;

    const float* arow = agg2 + (size_t)(row0 + m) * GHID;
    for (int nt = 0; nt < 4; ++nt) {
        const int col = nt * 16 + n;
        float bias = b3[col];
        v8f c;
        for (int r = 0; r < 8; ++r) c[r] = bias;
        for (int k = 0; k < GHID; k += 4) {
            const int ka = k + 2 * half;
            v2f a, b;
            a[0] = arow[ka];
            a[1] = arow[ka + 1];
            b[0] = W3[(size_t)ka * GOUT_DIM + col];
            b[1] = W3[(size_t)(ka + 1) * GOUT_DIM + col];
            c = __builtin_amdgcn_wmma_f32_16x16x4_f32(false, a, false, b,
                                                      (short)0, c, false, false);
        }
        for (int r = 0; r < 8; ++r)
            out[(size_t)(row0 + r + 8 * half) * GOUT_DIM + col] = fmaxf(c[r], 0.0f);
    }
}

// ---------------------------------------------------------------------------
extern "C" void kernel_launch(void* const* d_in, const int* in_sizes, int n_in,
                              void* d_out, int out_size, void* d_ws, size_t ws_size,
                              hipStream_t stream) {
    const float* x   = (const float*)d_in[0];
    const int*   ei  = (const int*)d_in[1];          // [2, E] int32 (JAX x64 off)
    const float* W1  = (const float*)d_in[2];
    const float* b1  = (const float*)d_in[3];
    const float* W2  = (const float*)d_in[4];
    const float* b2  = (const float*)d_in[5];
    const float* W3  = (const float*)d_in[6];
    const float* b3  = (const float*)d_in[7];
    float*       out = (float*)d_out;

    const int* src = ei;
    const int* dst = ei + GN_EDGES;

    // Workspace layout (fp32): agg1[N*64] | h[N*32] | agg2[N*32]  = 51.2 MB
    float* agg1 = (float*)d_ws;
    float* hbuf = agg1 + (size_t)GN_NODES * GIN_DIM;
    float* agg2 = hbuf + (size_t)GN_NODES * GHID;

    // 1) agg1 = x
    gin_copy64<<<(GN_NODES * GIN_DIM / 4 + 255) / 256, 256, 0, stream>>>(x, agg1);
    // 2) agg1 += scatter(x[src] -> dst)
    gin_scatter64<<<(GN_EDGES * 16 + 255) / 256, 256, 0, stream>>>(x, src, dst, agg1);
    // 3) h = relu(relu(agg1@W1+b1)@W2+b2); agg2 = h
    gin_mlp1<<<(GN_TILES + 7) / 8, 256, 0, stream>>>(agg1, W1, b1, W2, b2, hbuf, agg2);
    // 4) agg2 += scatter(h[src] -> dst)
    gin_scatter32<<<(GN_EDGES * 8 + 255) / 256, 256, 0, stream>>>(hbuf, src, dst, agg2);
    // 5) out = relu(agg2@W3+b3)
    gin_mlp2<<<(GN_TILES + 7) / 8, 256, 0, stream>>>(agg2, W3, b3, out);
}